// HANLogModel_27255862460871
// MI455X (gfx1250) — compile-verified
//
#include <hip/hip_runtime.h>

typedef float v8f __attribute__((ext_vector_type(8)));
typedef float v2f __attribute__((ext_vector_type(2)));

#define T_   27
#define N_   16384
#define B_   256
#define IN_  300
#define HID_ 128
#define OUT_ 64
#define ROWS_PER_BLK 64

// ---------------------------------------------------------------------------
// Kernel 0: zero the pooled-sum / count workspace (harness poisons ws).
// ---------------------------------------------------------------------------
__global__ void zero_ws_kernel(float* __restrict__ ws, int n) {
    int i = blockIdx.x * blockDim.x + threadIdx.x;
    int stride = gridDim.x * blockDim.x;
    for (; i < n; i += stride) ws[i] = 0.0f;
}

// ---------------------------------------------------------------------------
// Kernel 1: segment-sum pooling. segment_ids are sorted per type, so each
// 64-row block accumulates runs in registers and flushes one atomicAdd per
// segment boundary per column (~2 flushes/block instead of 64).
// ---------------------------------------------------------------------------
__global__ void pool_accum_kernel(const float* __restrict__ feat,
                                  const int*   __restrict__ seg,
                                  float* __restrict__ sums,   // [T,B,IN]
                                  float* __restrict__ cnts)   // [T,B]
{
    const int blk = blockIdx.x;
    const int t   = blk / (N_ / ROWS_PER_BLK);
    const int g   = blk % (N_ / ROWS_PER_BLK);
    const int tid = threadIdx.x;

    const int  c0   = tid;         // always < 300
    const int  c1   = tid + 256;   // valid only for tid < 44
    const bool has1 = (c1 < IN_);

    const float* frow   = feat + ((size_t)t * N_ + (size_t)g * ROWS_PER_BLK) * IN_;
    const int*   segrow = seg + t * N_ + g * ROWS_PER_BLK;

    float a0 = 0.0f, a1 = 0.0f;
    int cur = segrow[0];
    int run = 0;

    for (int r = 0; r < ROWS_PER_BLK; ++r) {
        const int s = segrow[r];
        if (s != cur) {
            float* dst = sums + ((size_t)t * B_ + cur) * IN_;
            atomicAdd(dst + c0, a0);
            if (has1) atomicAdd(dst + c1, a1);
            if (tid == 0) atomicAdd(cnts + t * B_ + cur, (float)run);
            a0 = 0.0f; a1 = 0.0f; run = 0; cur = s;
        }
        const float* fp = frow + (size_t)r * IN_;
        __builtin_prefetch(fp + IN_, 0, 0);   // global_prefetch_b8 for next row
        a0 += fp[c0];
        if (has1) a1 += fp[c1];
        ++run;
    }
    float* dst = sums + ((size_t)t * B_ + cur) * IN_;
    atomicAdd(dst + c0, a0);
    if (has1) atomicAdd(dst + c1, a1);
    if (tid == 0) atomicAdd(cnts + t * B_ + cur, (float)run);
}

// ---------------------------------------------------------------------------
// Kernel 2: per-type 2-layer MLP using V_WMMA_F32_16X16X4_F32.
// One 256-thread block (8 wave32) per type.
// LDS: W1 tile (150 KB) + h (128 KB) + inv-counts (1 KB) ~= 286 KB dynamic.
// ---------------------------------------------------------------------------
extern __shared__ float smem[];

__global__ void mlp_wmma_kernel(const float* __restrict__ sums,
                                const float* __restrict__ cnts,
                                const float* __restrict__ W1,
                                const float* __restrict__ b1,
                                const float* __restrict__ W2,
                                const float* __restrict__ b2,
                                float* __restrict__ out)   // [B, T, OUT]
{
    const int t     = blockIdx.x;
    const int tid   = threadIdx.x;
    const int wave  = tid >> 5;
    const int lane  = tid & 31;
    const int lhalf = lane >> 4;   // 0 or 1 -> selects K pair / M half
    const int lmod  = lane & 15;

    float* lds_w1  = smem;                    // IN_*HID_ floats
    float* lds_h   = smem + IN_ * HID_;       // B_*HID_ floats
    float* lds_inv = lds_h + B_ * HID_;       // B_ floats

    // inverse counts (segment mean, empty segments -> /1 -> 0)
    if (tid < B_) {
        float c = cnts[t * B_ + tid];
        lds_inv[tid] = 1.0f / fmaxf(c, 1.0f);
    }
    // stage W1[t] into LDS with float4 loads
    {
        const float4* src = reinterpret_cast<const float4*>(W1 + (size_t)t * IN_ * HID_);
        float4* dst = reinterpret_cast<float4*>(lds_w1);
        for (int i = tid; i < (IN_ * HID_) / 4; i += 256) dst[i] = src[i];
    }
    __syncthreads();

    const v8f vzero = {0.f, 0.f, 0.f, 0.f, 0.f, 0.f, 0.f, 0.f};

    // ---------------- Layer 1: pooled[256x300] x W1[300x128] -> h ----------
    const float* Abase = sums + (size_t)t * B_ * IN_;
    for (int mi = 0; mi < 2; ++mi) {
        const int   mt   = wave + mi * 8;          // M tile 0..15
        const int   arow = mt * 16 + lmod;         // this lane's A row
        const float minv = lds_inv[arow];

        v8f acc[8];
        #pragma unroll
        for (int nt = 0; nt < 8; ++nt) acc[nt] = vzero;

        for (int k = 0; k < IN_; k += 4) {
            const int kb = k + lhalf * 2;
            const float2 av = *reinterpret_cast<const float2*>(
                Abase + (size_t)arow * IN_ + kb);
            v2f a; a.x = av.x * minv; a.y = av.y * minv;
            #pragma unroll
            for (int nt = 0; nt < 8; ++nt) {
                const int n = nt * 16 + lmod;
                v2f b;
                b.x = lds_w1[kb * HID_ + n];
                b.y = lds_w1[(kb + 1) * HID_ + n];
                acc[nt] = __builtin_amdgcn_wmma_f32_16x16x4_f32(
                    false, a, false, b, (short)0, acc[nt], false, false);
            }
        }
        #pragma unroll
        for (int nt = 0; nt < 8; ++nt) {
            const int   n    = nt * 16 + lmod;
            const float bias = b1[t * HID_ + n];
            #pragma unroll
            for (int v = 0; v < 8; ++v) {
                const int m = mt * 16 + v + lhalf * 8;
                lds_h[m * HID_ + n] = fmaxf(acc[nt][v] + bias, 0.0f);
            }
        }
    }
    __syncthreads();

    // ---------------- Layer 2: h[256x128] x W2[128x64] -> out --------------
    const float* W2t = W2 + (size_t)t * HID_ * OUT_;
    for (int mi = 0; mi < 2; ++mi) {
        const int mt   = wave + mi * 8;
        const int arow = mt * 16 + lmod;

        v8f acc[4];
        #pragma unroll
        for (int nt = 0; nt < 4; ++nt) acc[nt] = vzero;

        for (int k = 0; k < HID_; k += 4) {
            const int kb = k + lhalf * 2;
            const float2 av = *reinterpret_cast<const float2*>(
                lds_h + arow * HID_ + kb);        // ds_load_b64
            v2f a; a.x = av.x; a.y = av.y;
            #pragma unroll
            for (int nt = 0; nt < 4; ++nt) {
                const int n = nt * 16 + lmod;
                v2f b;
                b.x = W2t[kb * OUT_ + n];
                b.y = W2t[(kb + 1) * OUT_ + n];
                acc[nt] = __builtin_amdgcn_wmma_f32_16x16x4_f32(
                    false, a, false, b, (short)0, acc[nt], false, false);
            }
        }
        #pragma unroll
        for (int nt = 0; nt < 4; ++nt) {
            const int   n    = nt * 16 + lmod;
            const float bias = b2[t * OUT_ + n];
            #pragma unroll
            for (int v = 0; v < 8; ++v) {
                const int m = mt * 16 + v + lhalf * 8;   // batch index
                out[(size_t)m * (T_ * OUT_) + t * OUT_ + n] = acc[nt][v] + bias;
            }
        }
    }
}

// ---------------------------------------------------------------------------
extern "C" void kernel_launch(void* const* d_in, const int* in_sizes, int n_in,
                              void* d_out, int out_size, void* d_ws, size_t ws_size,
                              hipStream_t stream) {
    const float* feat = (const float*)d_in[0];
    const int*   seg  = (const int*)d_in[1];
    const float* W1   = (const float*)d_in[2];
    const float* b1   = (const float*)d_in[3];
    const float* W2   = (const float*)d_in[4];
    const float* b2   = (const float*)d_in[5];

    float* sums = (float*)d_ws;                         // [T,B,IN]
    float* cnts = sums + (size_t)T_ * B_ * IN_;         // [T,B]

    const int nzero = T_ * B_ * IN_ + T_ * B_;
    zero_ws_kernel<<<1024, 256, 0, stream>>>(sums, nzero);

    pool_accum_kernel<<<T_ * (N_ / ROWS_PER_BLK), 256, 0, stream>>>(
        feat, seg, sums, cnts);

    const size_t smem_bytes = (size_t)(IN_ * HID_ + B_ * HID_ + B_) * sizeof(float);
    mlp_wmma_kernel<<<T_, 256, smem_bytes, stream>>>(
        sums, cnts, W1, b1, W2, b2, (float*)d_out);
}